// QML_hybrid_pretrained_graph_76828374991617
// MI455X (gfx1250) — compile-verified
//
#include <hip/hip_runtime.h>
#include <hip/hip_bf16.h>
#include <cmath>

// ---------------------------------------------------------------------------
// Sizes fixed by the reference
// ---------------------------------------------------------------------------
#define BATCH   16384
#define D_IN    256
#define D_H     512
#define NQ      12
#define NL      4
#define NSTATE  4096          // 2^12
#define SIM_T   256           // 8 wave32s per workgroup

typedef float v2f __attribute__((ext_vector_type(2)));
typedef float v8f __attribute__((ext_vector_type(8)));

// ---------------------------------------------------------------------------
// Kernel 1: h = relu(data @ W_graph + b_graph) via V_WMMA_F32_16X16X4_F32.
//
// Each wave owns a 64x16 strip of h (4 stacked 16x16 tiles) so one B
// fragment (the expensive stride-2KB loads through W_graph) feeds 4 WMMAs:
// VMEM per WMMA drops from ~3 to ~1.5 and strided-B traffic drops 4x.
//
// Fragment layouts (CDNA5 ISA 7.12.2, wave32):
//  A 16x4 f32 : lanes 0-15 row M=lane hold {K,K+1}; lanes 16-31 hold {K+2,K+3}
//  B 4x16 f32 : lanes 0-15 col N=lane hold {K,K+1}; lanes 16-31 hold {K+2,K+3}
//  C/D 16x16  : VGPR r -> row base + r + 8*(lane>=16), col n0 + (lane&15)
// ---------------------------------------------------------------------------
__global__ __launch_bounds__(256)
void gemm1_relu_wmma(const float* __restrict__ A,      // (BATCH, D_IN)
                     const float* __restrict__ W,      // (D_IN, D_H)
                     const float* __restrict__ bias,   // (D_H)
                     float* __restrict__ H)            // (BATCH, D_H)
{
    const int lane = threadIdx.x & 31;
    const int wave = threadIdx.x >> 5;
    const int id   = blockIdx.x * 8 + wave;   // (BATCH/64)*(D_H/16) = 8192 wave jobs
    const int mb   = id >> 5;                 // 64-row block index (0..255)
    const int nt   = id & 31;                 // 16-col tile index  (0..31)
    const int m0   = mb * 64;
    const int n0   = nt * 16;

    const int row = lane & 15;                // also the B column within the tile
    const int kb  = (lane >> 4) << 1;         // 0 for lanes 0-15, 2 for lanes 16-31

    const float* arow = A + (size_t)(m0 + row) * D_IN + kb;   // {k,k+1} contiguous
    const float* bcol = W + (size_t)kb * D_H + n0 + row;      // {k,k+1} stride D_H

    v8f acc[4] = {};
    #pragma unroll 4
    for (int k = 0; k < D_IN; k += 4) {
        v2f b;
        b.x = bcol[(size_t)k * D_H];
        b.y = bcol[(size_t)(k + 1) * D_H];
        #pragma unroll
        for (int t = 0; t < 4; ++t) {
            v2f a = *(const v2f*)(arow + (size_t)t * 16 * D_IN + k);  // 8B aligned
            acc[t] = __builtin_amdgcn_wmma_f32_16x16x4_f32(
                /*neg_a=*/false, a, /*neg_b=*/false, b,
                /*c_mod=*/(short)0, acc[t], /*reuse_a=*/false, /*reuse_b=*/false);
        }
    }

    const float bv   = bias[n0 + row];
    const int   mofs = (lane >> 4) << 3;      // +8 rows for upper half-wave
    #pragma unroll
    for (int t = 0; t < 4; ++t) {
        #pragma unroll
        for (int r = 0; r < 8; ++r) {
            float v = acc[t][r] + bv;
            H[(size_t)(m0 + t * 16 + r + mofs) * D_H + n0 + row] = v > 0.f ? v : 0.f;
        }
    }
}

// ---------------------------------------------------------------------------
// RY on wire w (reference convention: paired bit is bit (11-w), i.e. stride
// 2^(11-w)).  Pairs are disjoint per thread -> no intra-gate hazard.
// ---------------------------------------------------------------------------
__device__ __forceinline__
void apply_ry(float* st, int wire, float c, float s, int tid)
{
    const int stride = 1 << (11 - wire);
    #pragma unroll 4
    for (int p = tid; p < NSTATE / 2; p += SIM_T) {
        const int i0 = ((p & ~(stride - 1)) << 1) | (p & (stride - 1));
        const int i1 = i0 + stride;
        const float a = st[i0];
        const float b = st[i1];
        st[i0] = c * a - s * b;
        st[i1] = s * a + c * b;
    }
}

// ---------------------------------------------------------------------------
// Kernel 2: fused  x = h @ W_pre + b_pre  ->  angles  ->  12-qubit statevector
// simulation in LDS  ->  expectation value.  One workgroup per batch row.
// LDS: 16 KB state + angle tables  (~20 concurrent WGs per 320 KB WGP).
// ---------------------------------------------------------------------------
__global__ __launch_bounds__(SIM_T)
void qsim_kernel(const float* __restrict__ H,     // (BATCH, D_H)
                 const float* __restrict__ Wpre,  // (D_H, NQ)
                 const float* __restrict__ bpre,  // (NQ)
                 const float* __restrict__ qw,    // (NL*NQ)
                 float* __restrict__ out)         // (BATCH)
{
    __shared__ float st[NSTATE];
    __shared__ float cang[NQ],      sang[NQ];        // data-encoding angles
    __shared__ float qc[NL * NQ],   qs[NL * NQ];     // variational angles
    __shared__ float ssum;

    const int tid = threadIdx.x;
    const int b   = blockIdx.x;

    // ---- wave 0: 12-way dot product of h-row with W_pre, then angles -------
    if (tid < 32) {
        float acc[NQ];
        #pragma unroll
        for (int j = 0; j < NQ; ++j) acc[j] = 0.f;
        for (int k = tid; k < D_H; k += 32) {
            const float hv = H[(size_t)b * D_H + k];
            #pragma unroll
            for (int j = 0; j < NQ; ++j) acc[j] += hv * Wpre[k * NQ + j];
        }
        #pragma unroll
        for (int j = 0; j < NQ; ++j) {
            float v = acc[j];
            for (int off = 16; off > 0; off >>= 1) v += __shfl_down(v, off, 32);
            if (tid == 0) {
                const float xv = v + bpre[j];
                const float th = atanf(xv) + 1.5707963267948966f;
                cang[j] = cosf(0.5f * th);
                sang[j] = sinf(0.5f * th);
            }
        }
    }
    // ---- precompute variational cos/sin once (threads 32..79) --------------
    if (tid >= 32 && tid < 32 + NL * NQ) {
        const int   j  = tid - 32;
        const float th = 0.5f * qw[j];
        qc[j] = cosf(th);
        qs[j] = sinf(th);
    }
    if (tid == SIM_T - 1) ssum = 0.f;

    // ---- |0...0> init ------------------------------------------------------
    for (int k = tid; k < NSTATE; k += SIM_T) st[k] = (k == 0) ? 1.f : 0.f;
    __syncthreads();

    // ---- data-encoding RY layer (per-batch angles) -------------------------
    for (int w = 0; w < NQ; ++w) {
        apply_ry(st, w, cang[w], sang[w], tid);
        __syncthreads();
    }

    // ---- variational layers: shared RYs + CNOT ladder ----------------------
    for (int layer = 0; layer < NL; ++layer) {
        for (int w = 0; w < NQ; ++w) {
            const int g = layer * NQ + w;
            apply_ry(st, w, qc[g], qs[g], tid);
            __syncthreads();
        }
        for (int w = 0; w < NQ - 1; ++w) {
            // control bit = 11-w, target bit = 10-w; swap target where ctl==1
            const int tb = 1 << (10 - w);
            const int cb = tb << 1;
            #pragma unroll 2
            for (int q = tid; q < NSTATE / 4; q += SIM_T) {
                const int idx = ((q & ~(tb - 1)) << 2) | cb | (q & (tb - 1));
                const float a  = st[idx];
                const float bb = st[idx | tb];
                st[idx]      = bb;
                st[idx | tb] = a;
            }
            __syncthreads();
        }
    }

    // ---- <Z-sum> expectation: coeff[k] = 12 - 2*popcount(k) ----------------
    float local = 0.f;
    for (int k = tid; k < NSTATE; k += SIM_T) {        // stride-256: bank-conflict free
        const float v = st[k];
        local += v * v * (float)(NQ - 2 * __popc(k));
    }
    atomicAdd(&ssum, local);                           // ds_add_f32
    __syncthreads();
    if (tid == 0) out[b] = ssum;
}

// ---------------------------------------------------------------------------
// Launcher. d_ws holds h (BATCH * D_H * 4 = 32 MB).
// ---------------------------------------------------------------------------
extern "C" void kernel_launch(void* const* d_in, const int* in_sizes, int n_in,
                              void* d_out, int out_size, void* d_ws, size_t ws_size,
                              hipStream_t stream)
{
    const float* data = (const float*)d_in[0];
    const float* Wg   = (const float*)d_in[1];
    const float* bg   = (const float*)d_in[2];
    const float* Wp   = (const float*)d_in[3];
    const float* bp   = (const float*)d_in[4];
    const float* qw   = (const float*)d_in[5];

    float* H   = (float*)d_ws;
    float* out = (float*)d_out;

    // (16384/64)*(512/16) = 8192 wave jobs, 8 waves per 256-thread block
    gemm1_relu_wmma<<<8192 / 8, 256, 0, stream>>>(data, Wg, bg, H);
    qsim_kernel<<<BATCH, SIM_T, 0, stream>>>(H, Wp, bp, qw, out);
}